// loadNodes_1322849927756
// MI455X (gfx1250) — compile-verified
//
#include <hip/hip_runtime.h>

// Problem constants from the reference
#define ROWS_C   4194304   // 2*L*M*F
#define NODES_C  524288    // 2*L*M
#define NNZ_C    8388608

typedef float v2f __attribute__((ext_vector_type(2)));
typedef float v4f __attribute__((ext_vector_type(4)));
typedef float v8f __attribute__((ext_vector_type(8)));
typedef int   v4i __attribute__((ext_vector_type(4)));

// ---------------------------------------------------------------------------
// Zero the two segment-sum accumulators (o and lw), 2*ROWS floats, b128 stores
// (regular temporal stores: these lines are about to be hammered by L2 atomics)
// ---------------------------------------------------------------------------
__global__ __launch_bounds__(256) void zero_ws_kernel(v4f* __restrict__ p, int n4) {
    int i = blockIdx.x * blockDim.x + threadIdx.x;
    if (i < n4) p[i] = (v4f){0.f, 0.f, 0.f, 0.f};
}

// ---------------------------------------------------------------------------
// Sorted-segment scatter: acc[row[i]] += vals[i] * src[gath[i]]
// Each thread owns 8 consecutive nnz (rows sorted) -> register-accumulate runs
// of equal row, one hardware f32 atomic per run boundary.
// Streamed arrays (vals/rows/gath) are read exactly once -> non-temporal loads
// so the 16MB gather table and 16MB accumulator stay L2-resident.
// ---------------------------------------------------------------------------
__device__ __forceinline__ void atomic_fadd(float* p, float v) {
    __hip_atomic_fetch_add(p, v, __ATOMIC_RELAXED, __HIP_MEMORY_SCOPE_AGENT);
}

__global__ __launch_bounds__(256) void scatter_kernel(
    const float* __restrict__ vals, const int* __restrict__ rows,
    const int* __restrict__ gath, const float* __restrict__ src,
    float* __restrict__ acc)
{
    long t = (long)blockIdx.x * blockDim.x + threadIdx.x;
    long base = t * 8;

    v4f v0 = __builtin_nontemporal_load((const v4f*)(vals + base));
    v4f v1 = __builtin_nontemporal_load((const v4f*)(vals + base + 4));
    v4i r0 = __builtin_nontemporal_load((const v4i*)(rows + base));
    v4i r1 = __builtin_nontemporal_load((const v4i*)(rows + base + 4));
    v4i g0 = __builtin_nontemporal_load((const v4i*)(gath + base));
    v4i g1 = __builtin_nontemporal_load((const v4i*)(gath + base + 4));

    float m[8];
    int   r[8];
#pragma unroll
    for (int k = 0; k < 4; ++k) {
        m[k]     = v0[k] * src[g0[k]];  r[k]     = r0[k];
        m[k + 4] = v1[k] * src[g1[k]];  r[k + 4] = r1[k];
    }

    float a = m[0];
    int   cur = r[0];
#pragma unroll
    for (int k = 1; k < 8; ++k) {
        if (r[k] == cur) {
            a += m[k];
        } else {
            atomic_fadd(acc + cur, a);
            cur = r[k];
            a = m[k];
        }
    }
    atomic_fadd(acc + cur, a);
}

// ---------------------------------------------------------------------------
// F-reduction via V_WMMA_F32_16X16X4_F32.
// Per wave: 16 nodes. A[m,k] = (x*o)[node m, f-slice], B = ones(4x16)
// => D[m,n] = sum_f x[m,f]*o[m,f] in every column n.
// Two accumulating WMMAs cover f=0..3 and f=4..7.
// A layout (32-bit 16x4): lanes 0-15 & 16-31 both hold M=0..15; the two VGPRs
// per lane hold distinct K values -> each lane contributes 2 of the 8 f's:
//   lane<16 : f = {0,1} (+4 for second WMMA);  lane>=16 : f = {2,3} (+4).
// D layout: VGPR r = row r (lanes 0-15) and row r+8 (lanes 16-31).
// ADD_WL=false : store rowsum (pass-1 weightLoad)
// ADD_WL=true  : store wl_add[n] + rowsum (final output)
// ---------------------------------------------------------------------------
template <bool ADD_WL>
__global__ __launch_bounds__(256) void reduce_wmma_kernel(
    const float* __restrict__ x, const float* __restrict__ o,
    const float* __restrict__ wl_add, float* __restrict__ out)
{
    int gwave = (int)((blockIdx.x * blockDim.x + threadIdx.x) >> 5);
    int lane  = threadIdx.x & 31;
    int m     = lane & 15;
    int hi    = lane >> 4;                 // 0: K=0,1  |  1: K=2,3

    long nodeBase = (long)gwave * 16;      // 16 nodes per wave
    long off = (nodeBase + m) * 8 + hi * 2;

    v2f x1 = *(const v2f*)(x + off);
    v2f o1 = *(const v2f*)(o + off);
    v2f x2 = *(const v2f*)(x + off + 4);
    v2f o2 = *(const v2f*)(o + off + 4);

    v2f a1 = x1 * o1;                      // elementwise product, f 0..3 slice
    v2f a2 = x2 * o2;                      // f 4..7 slice
    v2f b  = {1.0f, 1.0f};                 // B = all ones (layout-invariant)
    v8f c  = {};

    c = __builtin_amdgcn_wmma_f32_16x16x4_f32(false, a1, false, b, (short)0, c, false, false);
    c = __builtin_amdgcn_wmma_f32_16x16x4_f32(false, a2, false, b, (short)0, c, false, false);

    // lane 0 holds rows 0..7 in c[0..7]; lane 16 holds rows 8..15 in c[0..7]
    if (m == 0) {
        long dbase = nodeBase + hi * 8;
        v4f lo, hv;
        if (ADD_WL) {
            v4f w0 = *(const v4f*)(wl_add + dbase);
            v4f w1 = *(const v4f*)(wl_add + dbase + 4);
            lo = (v4f){w0[0] + c[0], w0[1] + c[1], w0[2] + c[2], w0[3] + c[3]};
            hv = (v4f){w1[0] + c[4], w1[1] + c[5], w1[2] + c[6], w1[3] + c[7]};
        } else {
            lo = (v4f){c[0], c[1], c[2], c[3]};
            hv = (v4f){c[4], c[5], c[6], c[7]};
        }
        *(v4f*)(out + dbase)     = lo;
        *(v4f*)(out + dbase + 4) = hv;
    }
}

// ---------------------------------------------------------------------------
// Orchestration
// ---------------------------------------------------------------------------
extern "C" void kernel_launch(void* const* d_in, const int* in_sizes, int n_in,
                              void* d_out, int out_size, void* d_ws, size_t ws_size,
                              hipStream_t stream) {
    const float* weight        = (const float*)d_in[0];  // [ROWS] flat
    const float* load          = (const float*)d_in[1];  // [ROWS] flat
    const float* fanout_vals   = (const float*)d_in[2];  // [NNZ]
    const float* wire_vals     = (const float*)d_in[3];  // [NNZ]
    const int*   fanout_row    = (const int*)d_in[4];    // [NNZ] sorted
    const int*   fanout_gather = (const int*)d_in[5];    // [NNZ] -> [0,ROWS)
    const int*   wire_row      = (const int*)d_in[6];    // [NNZ] sorted
    const int*   wire_gather   = (const int*)d_in[7];    // [NNZ] -> [0,NODES)
    float*       out           = (float*)d_out;          // [NODES]

    // Workspace layout: o[ROWS] | lw[ROWS] | weightLoad[NODES]
    float* o  = (float*)d_ws;
    float* lw = o + ROWS_C;
    float* wl = lw + ROWS_C;

    // 1) zero both accumulators (2*ROWS floats)
    {
        int n4 = (2 * ROWS_C) / 4;
        zero_ws_kernel<<<n4 / 256, 256, 0, stream>>>((v4f*)d_ws, n4);
    }
    // 2) pass-1 scatter: o[fr[i]] += fv[i] * weight[fg[i]]
    scatter_kernel<<<NNZ_C / 8 / 256, 256, 0, stream>>>(
        fanout_vals, fanout_row, fanout_gather, weight, o);
    // 3) weightLoad[n] = sum_f load[n,f]*o[n,f]   (WMMA rowsum)
    reduce_wmma_kernel<false><<<(NODES_C / 16) * 32 / 256, 256, 0, stream>>>(
        load, o, nullptr, wl);
    // 4) pass-2 scatter: lw[wr[i]] += wv[i] * weightLoad[wg[i]]
    scatter_kernel<<<NNZ_C / 8 / 256, 256, 0, stream>>>(
        wire_vals, wire_row, wire_gather, wl, lw);
    // 5) out[n] = weightLoad[n] + sum_f lw[n,f]*o[n,f]   (WMMA rowsum + add)
    reduce_wmma_kernel<true><<<(NODES_C / 16) * 32 / 256, 256, 0, stream>>>(
        lw, o, wl, out);
}